// TransitionUp_54631984005450
// MI455X (gfx1250) — compile-verified
//
#include <hip/hip_runtime.h>
#include <hip/hip_bf16.h>

typedef __attribute__((ext_vector_type(2))) float v2f;
typedef __attribute__((ext_vector_type(8))) float v8f;

#define BATCH   2
#define NPTS    16384
#define SPTS    4096
#define IN_CH   256
#define OUT_CH  128

// ---------------------------------------------------------------------------
// C[M x 128] = A[M x K] @ W[K x 128] + bias, fp32 via V_WMMA_F32_16X16X4_F32.
// One wave computes a 16x32 output tile (two 16x16 WMMA accumulators sharing
// each A fragment). Layouts per CDNA5 ISA 7.12.2:
//   A 16x4 fp32 : lanes 0-15 -> M=lane, K = {0,1}; lanes 16-31 -> M=lane-16, K={2,3}
//   B 4x16 fp32 : lanes 0-15 -> N=lane, K = {0,1}; lanes 16-31 -> N=lane-16, K={2,3}
//   C/D 16x16   : VGPR r -> M=r (lanes 0-15) / M=r+8 (lanes 16-31), N = lane&15
// ---------------------------------------------------------------------------
__global__ void gemm_bias_wmma(const float* __restrict__ A,
                               const float* __restrict__ W,
                               const float* __restrict__ bias,
                               float* __restrict__ C,
                               int K) {
    const int NC = OUT_CH; // 128
    int wid   = blockIdx.x * (blockDim.x >> 5) + (threadIdx.x >> 5);
    int lane  = threadIdx.x & 31;
    int mtile = wid >> 2;        // 4 pairs of N-tiles cover 128 columns
    int npair = wid & 3;

    int lrow  = lane & 15;
    int khalf = (lane >> 4) << 1;           // 0 or 2

    const float* arow = A + (size_t)(mtile * 16 + lrow) * K + khalf;
    int n0 = npair * 32 + lrow;             // first N-tile column for this lane
    const float* wp = W + (size_t)khalf * NC + n0;

    v8f acc0 = {};
    v8f acc1 = {};

    for (int k = 0; k < K; k += 4) {
        v2f a = *(const v2f*)(arow + k);    // A[m][k+khalf .. +1], 8B aligned
        const float* wk = wp + (size_t)k * NC;
        v2f b0, b1f;
        b0[0]  = wk[0];       b0[1]  = wk[NC];        // W[k+khalf][n0],   W[k+khalf+1][n0]
        b1f[0] = wk[16];      b1f[1] = wk[NC + 16];   // second N-tile (+16 cols)
        acc0 = __builtin_amdgcn_wmma_f32_16x16x4_f32(
            false, a, false, b0,  (short)0, acc0, false, false);
        acc1 = __builtin_amdgcn_wmma_f32_16x16x4_f32(
            false, a, false, b1f, (short)0, acc1, false, false);
    }

    float bv0 = bias[n0];
    float bv1 = bias[n0 + 16];
    int rowbase = mtile * 16 + ((lane >> 4) << 3);   // +8 for upper half-wave
    float* crow = C + (size_t)rowbase * NC;
#pragma unroll
    for (int r = 0; r < 8; ++r) {
        crow[(size_t)r * NC + n0]      = acc0[r] + bv0;
        crow[(size_t)r * NC + n0 + 16] = acc1[r] + bv1;
    }
}

// ---------------------------------------------------------------------------
// 3-NN over S=4096 down points. down_xyz cached in LDS (48 KB << 320 KB/WGP);
// every lane reads the same candidate per iteration -> broadcast, no bank
// conflicts. Branchless top-3 insert keeps earliest index on ties (matches
// top_k tie-breaking). Writes idx[3] + normalized inverse-distance weights.
// ---------------------------------------------------------------------------
__global__ void knn3(const float* __restrict__ ori_xyzp,
                     const float* __restrict__ down_xyzp,
                     int* __restrict__ idx3,
                     float* __restrict__ w3) {
    __shared__ float sx[SPTS], sy[SPTS], sz[SPTS];
    int b = blockIdx.y;
    const float4* dp = (const float4*)(down_xyzp + (size_t)b * SPTS * 4);
    for (int j = threadIdx.x; j < SPTS; j += blockDim.x) {
        float4 v = dp[j];
        sx[j] = v.x; sy[j] = v.y; sz[j] = v.z;
    }
    __syncthreads();

    int n = blockIdx.x * blockDim.x + threadIdx.x;
    const float4 o = ((const float4*)(ori_xyzp + ((size_t)b * NPTS + n) * 4))[0];
    float ox = o.x, oy = o.y, oz = o.z;

    float d0 = 3.4e38f, d1 = 3.4e38f, d2 = 3.4e38f;
    int   i0 = 0, i1 = 0, i2 = 0;
#pragma unroll 4
    for (int j = 0; j < SPTS; ++j) {
        float dx = ox - sx[j], dy = oy - sy[j], dz = oz - sz[j];
        float d  = fmaf(dx, dx, fmaf(dy, dy, dz * dz));
        bool c0 = d < d0, c1 = d < d1, c2 = d < d2;
        d2 = c1 ? d1 : (c2 ? d : d2);
        i2 = c1 ? i1 : (c2 ? j : i2);
        d1 = c0 ? d0 : (c1 ? d : d1);
        i1 = c0 ? i0 : (c1 ? j : i1);
        d0 = c0 ? d  : d0;
        i0 = c0 ? j  : i0;
    }

    float r0 = 1.0f / (d0 + 1e-8f);
    float r1 = 1.0f / (d1 + 1e-8f);
    float r2 = 1.0f / (d2 + 1e-8f);
    float inv = 1.0f / (r0 + r1 + r2);
    size_t p = (size_t)b * NPTS + n;
    idx3[p * 3 + 0] = i0; idx3[p * 3 + 1] = i1; idx3[p * 3 + 2] = i2;
    w3[p * 3 + 0] = r0 * inv; w3[p * 3 + 1] = r1 * inv; w3[p * 3 + 2] = r2 * inv;
}

// ---------------------------------------------------------------------------
// out[b,n,c] += sum_k w_k * down_f[b, idx_k, c]. Thread per (point, channel):
// a wave covers 32 consecutive channels of one point, so the three gathered
// rows are coalesced 128B segments and idx/w loads are wave-uniform.
// ---------------------------------------------------------------------------
__global__ void interp_add(const float* __restrict__ downf,
                           const int* __restrict__ idx3,
                           const float* __restrict__ w3,
                           float* __restrict__ out) {
    size_t t = (size_t)blockIdx.x * blockDim.x + threadIdx.x;
    int    c = (int)(t & (OUT_CH - 1));
    size_t p = t >> 7;                 // point index in [0, B*N)
    int    b = (int)(p >> 14);         // / NPTS

    int   j0 = idx3[p * 3 + 0], j1 = idx3[p * 3 + 1], j2 = idx3[p * 3 + 2];
    float w0 = w3[p * 3 + 0],  w1 = w3[p * 3 + 1],  w2 = w3[p * 3 + 2];

    const float* dfb = downf + (size_t)b * SPTS * OUT_CH;
    float v = out[t];
    v = fmaf(w0, dfb[(size_t)j0 * OUT_CH + c], v);
    v = fmaf(w1, dfb[(size_t)j1 * OUT_CH + c], v);
    v = fmaf(w2, dfb[(size_t)j2 * OUT_CH + c], v);
    out[t] = v;
}

extern "C" void kernel_launch(void* const* d_in, const int* in_sizes, int n_in,
                              void* d_out, int out_size, void* d_ws, size_t ws_size,
                              hipStream_t stream) {
    const float* down_xyzp     = (const float*)d_in[0];
    const float* down_features = (const float*)d_in[1];
    const float* ori_xyzp      = (const float*)d_in[2];
    const float* ori_features  = (const float*)d_in[3];
    const float* W1 = (const float*)d_in[4];
    const float* b1 = (const float*)d_in[5];
    const float* W2 = (const float*)d_in[6];
    const float* b2 = (const float*)d_in[7];
    float* out = (float*)d_out;

    // workspace layout
    const size_t downf_elems = (size_t)BATCH * SPTS * OUT_CH;          // 1 Mi floats
    const size_t nidx_elems  = (size_t)BATCH * NPTS * 3;
    float* downf = (float*)d_ws;
    int*   idx3  = (int*)((char*)d_ws + downf_elems * sizeof(float));
    float* w3    = (float*)((char*)idx3 + nidx_elems * sizeof(int));

    // 1) down_f = down_features @ W1 + b1  (M = B*S = 8192, K = 256)
    {
        int Mtotal = BATCH * SPTS;
        int waves  = (Mtotal / 16) * 4;       // 4 N-tile pairs per m-tile
        gemm_bias_wmma<<<waves / 8, 256, 0, stream>>>(down_features, W1, b1, downf, IN_CH);
    }
    // 2) ori_f = ori_features @ W2 + b2 -> directly into d_out (M = B*N = 32768, K = 128)
    {
        int Mtotal = BATCH * NPTS;
        int waves  = (Mtotal / 16) * 4;
        gemm_bias_wmma<<<waves / 8, 256, 0, stream>>>(ori_features, W2, b2, out, OUT_CH);
    }
    // 3) 3-NN search
    {
        dim3 grid(NPTS / 256, BATCH);
        knn3<<<grid, 256, 0, stream>>>(ori_xyzp, down_xyzp, idx3, w3);
    }
    // 4) weighted gather + accumulate into out
    {
        size_t total = (size_t)BATCH * NPTS * OUT_CH;  // 4 Mi threads
        interp_add<<<(int)(total / 256), 256, 0, stream>>>(downf, idx3, w3, out);
    }
}